// RoIAwarePool3d_4183298146660
// MI455X (gfx1250) — compile-verified
//
#include <hip/hip_runtime.h>
#include <hip/hip_bf16.h>

typedef __attribute__((ext_vector_type(2))) float v2f;
typedef __attribute__((ext_vector_type(8))) float v8f;
typedef int v4i_vs __attribute__((vector_size(16)));
typedef __attribute__((address_space(1))) v4i_vs* gv4i_p;
typedef __attribute__((address_space(3))) v4i_vs* lv4i_p;

#define OUT_RES 12
#define NVOX    (OUT_RES * OUT_RES * OUT_RES)   // 1728
#define NROIS   128
#define CFEAT   64
#define WS_F4   640            // 5 * 128 float4 descriptor table

// ---------------------------------------------------------------- atomics ---
__device__ __forceinline__ void atomic_fmax_dev(float* addr, float val) {
    // GLOBAL_ATOMIC_MAX_NUM_F32, no-return form, device scope so the fixup
    // kernel (separate dispatch) observes results.
    asm volatile("global_atomic_max_num_f32 %0, %1, off scope:SCOPE_DEV"
                 :: "v"(addr), "v"(val) : "memory");
}

// ------------------------------------------------------------------- prep ---
// Build per-ROI descriptors:
//  B-columns for lx/ly/lz:  local = [px py pz 1] . Bcol
//  H = half dims, S = OUT/d (voxel inverse size)
__global__ void roipool_prep(const float* __restrict__ rois, float4* __restrict__ ws) {
    int r = threadIdx.x;
    if (r >= NROIS) return;
    const float* roi = rois + r * 7;
    float cx = roi[0], cy = roi[1], cz = roi[2];
    float dx = roi[3], dy = roi[4], dz = roi[5], rz = roi[6];
    float c = cosf(rz), s = sinf(rz);
    float czc = cz + 0.5f * dz;
    ws[0 * NROIS + r] = make_float4(  c,   s, 0.0f, -(c * cx + s * cy)); // lx
    ws[1 * NROIS + r] = make_float4( -s,   c, 0.0f,  (s * cx - c * cy)); // ly
    ws[2 * NROIS + r] = make_float4(0.0f, 0.0f, 1.0f, -czc);            // lz
    ws[3 * NROIS + r] = make_float4(0.5f * dx, 0.5f * dy, 0.5f * dz, 0.0f);
    ws[4 * NROIS + r] = make_float4((float)OUT_RES / dx, (float)OUT_RES / dy,
                                    (float)OUT_RES / dz, 0.0f);
}

// ------------------------------------------------------------------- init ---
__global__ void roipool_init(float4* __restrict__ out, int n4) {
    int i = blockIdx.x * blockDim.x + threadIdx.x;
    if (i < n4) {
        float ninf = -__builtin_inff();
        out[i] = make_float4(ninf, ninf, ninf, ninf);
    }
}

// ------------------------------------------------------------------ fixup ---
__global__ void roipool_fixup(float4* __restrict__ out, int n4) {
    int i = blockIdx.x * blockDim.x + threadIdx.x;
    if (i >= n4) return;
    float ninf = -__builtin_inff();
    float4 v = out[i];
    v.x = (v.x == ninf) ? 0.0f : v.x;
    v.y = (v.y == ninf) ? 0.0f : v.y;
    v.z = (v.z == ninf) ? 0.0f : v.z;
    v.w = (v.w == ninf) ? 0.0f : v.w;
    out[i] = v;
}

// ------------------------------------------------------------------- pool ---
// One wave handles a tile of 16 points against all 128 ROIs.
// A (16x4 f32)  : rows = points, cols = (x, y, z, 1)
// B (4x16 f32)  : one axis for a group of 16 ROIs
// D (16x16 f32) : lane l holds axis value for roi g*16+(l&15),
//                 points tileBase + (l<16 ? 0..7 : 8..15) in VGPRs 0..7.
__launch_bounds__(256)
__global__ void roipool_pool(const float* __restrict__ pts,
                             const float* __restrict__ feat,
                             const float4* __restrict__ ws,
                             float* __restrict__ out, int npts) {
    __shared__ float4 lds[WS_F4];
    const int tid = threadIdx.x;

    // Stage descriptor table into LDS with CDNA5 async global->LDS DMA.
#if __has_builtin(__builtin_amdgcn_global_load_async_to_lds_b128)
    {
        float4* wsm = (float4*)ws;
        for (int i = tid; i < WS_F4; i += 256) {
            __builtin_amdgcn_global_load_async_to_lds_b128(
                (gv4i_p)(wsm + i),
                (lv4i_p)(&lds[i]),
                0, 0);
        }
#if __has_builtin(__builtin_amdgcn_s_wait_asynccnt)
        __builtin_amdgcn_s_wait_asynccnt(0);
#else
        asm volatile("s_wait_asynccnt 0" ::: "memory");
#endif
    }
#else
    for (int i = tid; i < WS_F4; i += 256) lds[i] = ws[i];
#endif
    __syncthreads();

    const int lane = tid & 31;
    const int half = lane >> 4;     // 0: lanes 0-15 (K=0,1 / M rows 0-7), 1: lanes 16-31
    const int l15  = lane & 15;
    const int wave = (blockIdx.x * 256 + tid) >> 5;
    const int tileBase = wave * 16;
    if (tileBase >= npts) return;   // wave-uniform: EXEC stays all-ones

    // ---- A matrix (clamp tail loads; bogus rows filtered in j-loop) ----
    int pa = tileBase + l15;
    if (pa > npts - 1) pa = npts - 1;
    float px = pts[3 * pa + 0], py = pts[3 * pa + 1], pz = pts[3 * pa + 2];
    v2f A;
    A.x = half ? pz   : px;
    A.y = half ? 1.0f : py;

    const float4* Bx = lds;
    const float4* By = lds + NROIS;
    const float4* Bz = lds + 2 * NROIS;
    const float4* Hh = lds + 3 * NROIS;
    const float4* Sc = lds + 4 * NROIS;

    const int pj0 = tileBase + half * 8;

    for (int g = 0; g < NROIS / 16; ++g) {
        const int roi = g * 16 + l15;
        float4 bx = Bx[roi], by = By[roi], bz = Bz[roi];
        v2f Ba, Bb, Bc;
        Ba.x = half ? bx.z : bx.x;  Ba.y = half ? bx.w : bx.y;
        Bb.x = half ? by.z : by.x;  Bb.y = half ? by.w : by.y;
        Bc.x = half ? bz.z : bz.x;  Bc.y = half ? bz.w : bz.y;

        v8f zero = {};
        v8f Dx = __builtin_amdgcn_wmma_f32_16x16x4_f32(false, A, false, Ba,
                                                       (short)0, zero, false, false);
        v8f Dy = __builtin_amdgcn_wmma_f32_16x16x4_f32(false, A, false, Bb,
                                                       (short)0, zero, false, false);
        v8f Dz = __builtin_amdgcn_wmma_f32_16x16x4_f32(false, A, false, Bc,
                                                       (short)0, zero, false, false);

        float4 h  = Hh[roi];
        float4 sc = Sc[roi];
        float* outroi = out + (size_t)roi * (NVOX * CFEAT);

#pragma unroll
        for (int j = 0; j < 8; ++j) {
            float lx = Dx[j], ly = Dy[j], lz = Dz[j];
            int pj = pj0 + j;
            bool inbox = (pj < npts) &&
                         (fabsf(lx) < h.x) && (fabsf(ly) < h.y) && (fabsf(lz) <= h.z);
            if (inbox) {
                int ix = (int)floorf((lx + h.x) * sc.x);
                int iy = (int)floorf((ly + h.y) * sc.y);
                int iz = (int)floorf((lz + h.z) * sc.z);
                ix = min(max(ix, 0), OUT_RES - 1);
                iy = min(max(iy, 0), OUT_RES - 1);
                iz = min(max(iz, 0), OUT_RES - 1);
                int vid = (ix * OUT_RES + iy) * OUT_RES + iz;
                float* o = outroi + vid * CFEAT;
                const float4* f = (const float4*)(feat + (size_t)pj * CFEAT);
#pragma unroll 1
                for (int c4 = 0; c4 < CFEAT / 4; ++c4) {
                    float4 v = f[c4];
                    atomic_fmax_dev(o + 4 * c4 + 0, v.x);
                    atomic_fmax_dev(o + 4 * c4 + 1, v.y);
                    atomic_fmax_dev(o + 4 * c4 + 2, v.z);
                    atomic_fmax_dev(o + 4 * c4 + 3, v.w);
                }
            }
        }
    }
}

// ----------------------------------------------------------------- launch ---
extern "C" void kernel_launch(void* const* d_in, const int* in_sizes, int n_in,
                              void* d_out, int out_size, void* d_ws, size_t ws_size,
                              hipStream_t stream) {
    const float* rois = (const float*)d_in[0];   // [128, 7]
    const float* pts  = (const float*)d_in[1];   // [P, 3]
    const float* feat = (const float*)d_in[2];   // [P, 64]
    float*  out = (float*)d_out;                 // [128, 12,12,12, 64]
    float4* ws  = (float4*)d_ws;

    const int npts = in_sizes[1] / 3;

    roipool_prep<<<1, 128, 0, stream>>>(rois, ws);

    const int n4 = out_size / 4;
    const int ib = (n4 + 255) / 256;
    roipool_init<<<ib, 256, 0, stream>>>((float4*)out, n4);

    const int waves  = (npts + 15) / 16;
    const int blocks = (waves + 7) / 8;          // 8 waves (256 thr) per block
    roipool_pool<<<blocks, 256, 0, stream>>>(pts, feat, ws, out, npts);

    roipool_fixup<<<ib, 256, 0, stream>>>((float4*)out, n4);
}